// ProposalFeatureExtractor_43508018708750
// MI455X (gfx1250) — compile-verified
//
#include <hip/hip_runtime.h>

// ROI-align (bilinear crop-resize). Bandwidth-bound: 202 MB of stores -> ~8.7us
// floor at 23.3 TB/s; source map (4.2 MB) lives in L2. CDNA5 path: Tensor Data
// Mover gather-mode descriptors stage the 56 needed source columns into LDS
// (tensor_load_to_lds, TENSORcnt), then a float2 blend with coalesced stores.

#define ROI 14
#define MAPW 64
#define CT 258
#define SLOT_STRIDE 260   // 256 data + TDM pad of 4 DWORDs per row

typedef unsigned int u32x4 __attribute__((ext_vector_type(4)));
typedef int          i32x4 __attribute__((ext_vector_type(4)));
typedef int          i32x8 __attribute__((ext_vector_type(8)));

__device__ __forceinline__ int iclamp(int v, int lo, int hi) {
  return v < lo ? lo : (v > hi ? hi : v);
}

__global__ __launch_bounds__(256)
void roi_align_tdm_kernel(const float* __restrict__ feat,
                          const float* __restrict__ pos,
                          const int* __restrict__ boxes,
                          float* __restrict__ out) {
  // 56 gathered columns (2 y-corners x 28 x-slots), TDM-padded stride 260 f32
  __shared__ __align__(16) float cols[56 * SLOT_STRIDE];   // 58,240 B
  __shared__ __align__(16) float posr[56 * 2];             // position channels

  const int b = blockIdx.x, oy = blockIdx.y, tid = threadIdx.x;
  const int y1 = boxes[4 * b + 0], x1 = boxes[4 * b + 1];
  const int y2 = boxes[4 * b + 2], x2 = boxes[4 * b + 3];
  const int h = y2 - y1, w = x2 - x1;

  // Exact integer floor/frac of sy = ((2*oy+1)*h - 14)/28 + y1
  const int   my  = (2 * oy + 1) * h - 14;
  const int   qy  = (my + 28) / 28 - 1;            // floor(my/28), my >= -28
  const float ty  = (float)(my - 28 * qy) * (1.0f / 28.0f);
  const int   yr0 = y1 + iclamp(qy,     0, h - 1);
  const int   yr1 = y1 + iclamp(qy + 1, 0, h - 1);

  auto xcol_of = [&](int ox, int j) {
    int mx = (2 * ox + 1) * w - 14;
    int qx = (mx + 28) / 28 - 1 + j;
    return x1 + iclamp(qx, 0, w - 1);
  };
  auto row_of = [&](int slot) {                    // slot in [0,56)
    int yc = (slot >= 28) ? 1 : 0;
    int xs = slot - 28 * yc;
    int yrow = yc ? yr1 : yr0;
    return yrow * MAPW + xcol_of(xs >> 1, xs & 1); // row in (4096, C) tensor
  };

  // ---- Stage via Tensor Data Mover gather-mode descriptors (wave 0 only) ----
  if (tid < 32) {
    const unsigned colsBase = (unsigned)(size_t)(void*)cols;
    const unsigned posrBase = (unsigned)(size_t)(void*)posr;
    const unsigned long long featGA = (unsigned long long)(size_t)feat;
    const unsigned long long posGA  = (unsigned long long)(size_t)pos;
    const i32x8 gzero = {0, 0, 0, 0, 0, 0, 0, 0};  // extra operand (clang-23 form)
    #pragma unroll
    for (int k = 0; k < 4; ++k) {                  // 4 descriptors x <=16 rows
      const int nIdx = (k < 3) ? 16 : 8;           // 16+16+16+8 = 56 rows
      int pk[8];
      #pragma unroll
      for (int d = 0; d < 8; ++d) {                // pack 16-bit row indices
        int s0 = 16 * k + 2 * d, s1 = s0 + 1;
        unsigned lo = (s0 < 56) ? (unsigned)row_of(s0) : 0u;
        unsigned hi = (s1 < 56) ? (unsigned)row_of(s1) : 0u;
        pk[d] = (int)(lo | (hi << 16));
      }
      { // feature tensor (4096 rows x 256 f32), pad 4 DWORDs/row -> stride 260
        u32x4 g0; i32x8 g1; i32x4 g2, g3;
        g0[0] = 1u | (1u << 31);                               // count=1, gather, 16b idx
        g0[1] = colsBase + (unsigned)(16 * k * SLOT_STRIDE * 4);
        g0[2] = (unsigned)(featGA & 0xffffffffull);
        g0[3] = (unsigned)((featGA >> 32) & 0x01ffffffull) | (2u << 30); // type=2
        g1[0] = (2 << 16) | (1 << 20) | (7 << 22) | (3 << 25); // 4B, pad: 4dw/256dw
        g1[1] = 256 << 16;                                     // tensor_dim0 = 256
        g1[2] = 4096 << 16;                                    // tensor_dim1 = 4096
        g1[3] = 256 << 16;                                     // tile_dim0 = 256
        g1[4] = nIdx;                                          // tile_dim1 = #indices
        g1[5] = 256;                                           // dim0_stride = 256
        g1[6] = 0; g1[7] = 0;
        g2[0] = pk[0]; g2[1] = pk[1]; g2[2] = pk[2]; g2[3] = pk[3];
        g3[0] = pk[4]; g3[1] = pk[5]; g3[2] = pk[6]; g3[3] = pk[7];
        __builtin_amdgcn_tensor_load_to_lds(g0, g1, g2, g3, gzero, 0);
      }
      { // position tensor (4096 rows x 2 f32), packed rows (8 B stride)
        u32x4 g0; i32x8 g1; i32x4 g2, g3;
        g0[0] = 1u | (1u << 31);
        g0[1] = posrBase + (unsigned)(16 * k * 2 * 4);
        g0[2] = (unsigned)(posGA & 0xffffffffull);
        g0[3] = (unsigned)((posGA >> 32) & 0x01ffffffull) | (2u << 30);
        g1[0] = (2 << 16);                                     // 4B, no pad
        g1[1] = 2 << 16;                                       // tensor_dim0 = 2
        g1[2] = 4096 << 16;                                    // tensor_dim1 = 4096
        g1[3] = 2 << 16;                                       // tile_dim0 = 2
        g1[4] = nIdx;
        g1[5] = 2;                                             // dim0_stride = 2
        g1[6] = 0; g1[7] = 0;
        g2[0] = pk[0]; g2[1] = pk[1]; g2[2] = pk[2]; g2[3] = pk[3];
        g3[0] = pk[4]; g3[1] = pk[5]; g3[2] = pk[6]; g3[3] = pk[7];
        __builtin_amdgcn_tensor_load_to_lds(g0, g1, g2, g3, gzero, 0);
      }
    }
    __builtin_amdgcn_s_wait_tensorcnt(0);          // drain TDM before barrier
  }
  __syncthreads();

  // ---- 4-corner blend out of LDS; float2 lanes -> coalesced b64 stores ----
  float* obase = out + (size_t)(b * ROI + oy) * (ROI * CT);
  const int npairs = (ROI * CT) / 2;               // 1806 float2 per block
  for (int p = tid; p < npairs; p += 256) {
    int ox = p / 129;                              // 129 pairs per ox (258 even)
    int cp = p - ox * 129;
    int c  = cp * 2;
    int mx = (2 * ox + 1) * w - 14;
    int qx = (mx + 28) / 28 - 1;
    float tx = (float)(mx - 28 * qx) * (1.0f / 28.0f);
    int s = 2 * ox;
    float2 A, B, C, D;
    if (c < 256) {
      A = *(const float2*)&cols[(s         ) * SLOT_STRIDE + c]; // (y0,x0)
      B = *(const float2*)&cols[(s + 1     ) * SLOT_STRIDE + c]; // (y0,xb)
      C = *(const float2*)&cols[(28 + s    ) * SLOT_STRIDE + c]; // (yb,x0)
      D = *(const float2*)&cols[(28 + s + 1) * SLOT_STRIDE + c]; // (yb,xb)
    } else {                                       // c == 256: position pair
      A = *(const float2*)&posr[(s         ) * 2];
      B = *(const float2*)&posr[(s + 1     ) * 2];
      C = *(const float2*)&posr[(28 + s    ) * 2];
      D = *(const float2*)&posr[(28 + s + 1) * 2];
    }
    float2 r;
    float topx = A.x + tx * (B.x - A.x);
    float botx = C.x + tx * (D.x - C.x);
    r.x = topx + ty * (botx - topx);
    float topy = A.y + tx * (B.y - A.y);
    float boty = C.y + tx * (D.y - C.y);
    r.y = topy + ty * (boty - topy);
    *(float2*)&obase[2 * p] = r;                   // contiguous across threads
  }
}

extern "C" void kernel_launch(void* const* d_in, const int* in_sizes, int n_in,
                              void* d_out, int out_size, void* d_ws, size_t ws_size,
                              hipStream_t stream) {
  const float* feat  = (const float*)d_in[0];   // (64,64,256) f32
  const float* pos   = (const float*)d_in[1];   // (64,64,2)   f32
  const int*   boxes = (const int*)d_in[2];     // (N,4)       i32
  float* out = (float*)d_out;                   // (N,14,14,258) f32
  int N = in_sizes[2] / 4;
  dim3 grid((unsigned)N, ROI);
  roi_align_tdm_kernel<<<grid, 256, 0, stream>>>(feat, pos, boxes, out);
}